// LogLikelihoodLayer_40389872451864
// MI455X (gfx1250) — compile-verified
//
#include <hip/hip_runtime.h>
#include <hip/hip_bf16.h>
#include <math.h>

typedef __attribute__((ext_vector_type(2))) float v2f;
typedef __attribute__((ext_vector_type(8))) float v8f;

#define DMAT 64
#define LDSS 68   // padded LDS row stride: 68*4 = 272 bytes = 16B-aligned rows, no bank conflicts

// One wave32 per (b,t) matrix: blocked Cholesky in LDS with fp32 WMMA Schur updates.
__global__ __launch_bounds__(32)
void gauss_nll_chol_kernel(const float* __restrict__ x,
                           const float* __restrict__ mu,
                           const float* __restrict__ sigma,
                           float* __restrict__ partials, int BT) {
    __shared__ float S[DMAT][LDSS];
    __shared__ float yv[DMAT];

    const int bid  = blockIdx.x;
    if (bid >= BT) return;
    const int lane = threadIdx.x;      // 0..31 (wave32)
    const int l15  = lane & 15;
    const int half = lane >> 4;        // 0 or 1

    // ---- Stage sigma tile (16 KB) into LDS via 128-bit loads/stores ----
    const float4* sg = (const float4*)(sigma + (size_t)bid * DMAT * DMAT);
    #pragma unroll 4
    for (int it = 0; it < 32; ++it) {
        int idx4 = it * 32 + lane;           // float4 index 0..1023
        int r = idx4 >> 4;                   // (idx4*4)/64
        int c = (idx4 & 15) << 2;            // (idx4*4)%64
        *(float4*)&S[r][c] = sg[idx4];
    }
    // d = x - mu
    for (int i = lane; i < DMAT; i += 32)
        yv[i] = x[(size_t)bid * DMAT + i] - mu[(size_t)bid * DMAT + i];
    __syncthreads();
    // jitter on diagonal (reference inverts sigma + 1e-8*I)
    for (int i = lane; i < DMAT; i += 32) S[i][i] += 1e-8f;
    __syncthreads();

    // ---- Blocked right-looking Cholesky, NB=16, 4 block steps ----
    for (int kb = 0; kb < 4; ++kb) {
        const int K = kb * 16;

        // 1) factor 16x16 diagonal block (lanes 0..15 own rows)
        for (int j = 0; j < 16; ++j) {
            if (lane == 0) S[K + j][K + j] = sqrtf(S[K + j][K + j]);
            __syncthreads();
            float ljj = S[K + j][K + j];
            if (lane > j && lane < 16)
                S[K + lane][K + j] /= ljj;
            __syncthreads();
            if (lane > j && lane < 16) {
                float lij = S[K + lane][K + j];
                for (int t = j + 1; t <= lane; ++t)
                    S[K + lane][K + t] -= lij * S[K + t][K + j];
            }
            __syncthreads();
        }

        // 2) panel triangular solve: rows below diag block, one row per lane
        for (int rr = K + 16 + lane; rr < DMAT; rr += 32) {
            #pragma unroll
            for (int j = 0; j < 16; ++j) {
                float t = S[rr][K + j];
                for (int p = 0; p < j; ++p)
                    t -= S[rr][K + p] * S[K + j][K + p];
                S[rr][K + j] = t / S[K + j][K + j];
            }
        }
        __syncthreads();

        // 3) trailing Schur update C -= L_ib * L_jb^T via V_WMMA_F32_16X16X4_F32
        for (int ib = kb + 1; ib < 4; ++ib) {
            for (int jb = kb + 1; jb <= ib; ++jb) {
                const int IB = ib * 16, JB = jb * 16;
                // C fragment: element (M = v + 8*half, N = l15)
                v8f c;
                #pragma unroll
                for (int v = 0; v < 8; ++v)
                    c[v] = S[IB + v + half * 8][JB + l15];
                #pragma unroll
                for (int k4 = 0; k4 < 16; k4 += 4) {
                    const int ac = K + k4 + half * 2;
                    v2f a, b;
                    // A fragment (negated): (M = l15, K = 2*half + {0,1})
                    a.x = -S[IB + l15][ac];
                    a.y = -S[IB + l15][ac + 1];
                    // B fragment = L_jb^T: (K = 2*half + {0,1}, N = l15)
                    b.x =  S[JB + l15][ac];
                    b.y =  S[JB + l15][ac + 1];
                    c = __builtin_amdgcn_wmma_f32_16x16x4_f32(
                            false, a, false, b, (short)0, c, false, false);
                }
                #pragma unroll
                for (int v = 0; v < 8; ++v)
                    S[IB + v + half * 8][JB + l15] = c[v];
            }
        }
        __syncthreads();
    }

    // ---- logdet = 2 * sum(log(L_ii)) ----
    float ld = 0.f;
    for (int i = lane; i < DMAT; i += 32) ld += __logf(S[i][i]);
    #pragma unroll
    for (int m = 16; m; m >>= 1) ld += __shfl_xor(ld, m, 32);
    ld *= 2.0f;

    // ---- forward substitution L y = d ; quad = ||y||^2 ----
    float quad = 0.f;
    for (int i = 0; i < DMAT; ++i) {
        float p = 0.f;
        for (int j = lane; j < i; j += 32) p += S[i][j] * yv[j];
        #pragma unroll
        for (int m = 16; m; m >>= 1) p += __shfl_xor(p, m, 32);
        if (lane == 0) {
            float yi = (yv[i] - p) / S[i][i];
            yv[i] = yi;
            quad += yi * yi;
        }
        __syncthreads();
    }

    if (lane == 0)
        partials[bid] = 0.5f * (ld + quad);   // per-matrix NLL contribution
}

// Deterministic final reduction (fixed order, no fp atomics).
__global__ __launch_bounds__(256)
void nll_reduce_kernel(const float* __restrict__ partials, float* __restrict__ out, int n) {
    __shared__ float sm[256];
    float acc = 0.f;
    for (int i = threadIdx.x; i < n; i += 256) acc += partials[i];
    sm[threadIdx.x] = acc;
    __syncthreads();
    for (int s = 128; s > 0; s >>= 1) {
        if ((int)threadIdx.x < s) sm[threadIdx.x] += sm[threadIdx.x + s];
        __syncthreads();
    }
    if (threadIdx.x == 0) out[0] = sm[0];
}

extern "C" void kernel_launch(void* const* d_in, const int* in_sizes, int n_in,
                              void* d_out, int out_size, void* d_ws, size_t ws_size,
                              hipStream_t stream) {
    const float* x     = (const float*)d_in[0];
    const float* mu    = (const float*)d_in[1];
    const float* sigma = (const float*)d_in[2];
    float* out = (float*)d_out;

    const int BT = in_sizes[0] / DMAT;       // 32*400 = 12800
    float* partials = (float*)d_ws;          // BT floats of scratch

    gauss_nll_chol_kernel<<<BT, 32, 0, stream>>>(x, mu, sigma, partials, BT);
    nll_reduce_kernel<<<1, 256, 0, stream>>>(partials, out, BT);
}